// LsqNonneg_56014963474995
// MI455X (gfx1250) — compile-verified
//
#include <hip/hip_runtime.h>
#include <math.h>

typedef float v2f __attribute__((ext_vector_type(2)));
typedef float v8f __attribute__((ext_vector_type(8)));

#define M_DIM 512
#define K_DIM 32
#define N_DIM 32768
#define FISTA_ITERS 200

// D = A(16x4 f32) * B(4x16 f32) + C(16x16 f32), wave32 WMMA.
__device__ __forceinline__ v8f wmma4(v2f a, v2f b, v8f c) {
  return __builtin_amdgcn_wmma_f32_16x16x4_f32(false, a, false, b, (short)0, c,
                                               false, false);
}

// ---------------------------------------------------------------------------
// Kernel 1 (1 wave): AtA = A^T A (32x32) via WMMA, then power iteration for
// L = ||AtA||_2. Writes ws[0..1023] = AtA row-major, ws[1024] = step = 1/L.
// ---------------------------------------------------------------------------
__global__ __launch_bounds__(32)
void nnls_prep_kernel(const float* __restrict__ A, float* __restrict__ ws) {
  __shared__ float ata[32][33];  // padded: conflict-free row reads
  const int lane = threadIdx.x;
  const int h = lane >> 4;       // lane-half (A/B layout: K selector)
  const int cl = lane & 15;      // column-in-tile

  v8f c00 = {}, c01 = {}, c10 = {}, c11 = {};
  for (int c = 0; c < M_DIM / 4; ++c) {
    const int m0 = 4 * c + 2 * h;
    // A-operand (== B-operand by identical index map) chunks of A^T / A.
    v2f aT, aB;
    aT.x = A[(m0 + 0) * K_DIM + cl];
    aT.y = A[(m0 + 1) * K_DIM + cl];
    aB.x = A[(m0 + 0) * K_DIM + 16 + cl];
    aB.y = A[(m0 + 1) * K_DIM + 16 + cl];
    c00 = wmma4(aT, aT, c00);
    c01 = wmma4(aT, aB, c01);
    c10 = wmma4(aB, aT, c10);
    c11 = wmma4(aB, aB, c11);
  }
  // C layout -> AtA[row][col] in LDS.
#pragma unroll
  for (int r = 0; r < 8; ++r) {
    ata[r + 8 * h][cl]           = c00[r];
    ata[r + 8 * h][16 + cl]      = c01[r];
    ata[16 + r + 8 * h][cl]      = c10[r];
    ata[16 + r + 8 * h][16 + cl] = c11[r];
  }
  __syncthreads();  // single-wave WG: lowers to S_NOP; dscnt ordering suffices

  float row[32];
#pragma unroll
  for (int j = 0; j < 32; ++j) row[j] = ata[lane][j];

  // Power iteration: AtA is SPD with a dominant eigenvalue, converges fast.
  float v = 1.0f, L = 1.0f;
  for (int it = 0; it < 64; ++it) {
    float w = 0.0f;
#pragma unroll
    for (int j = 0; j < 32; ++j) w = fmaf(row[j], __shfl(v, j, 32), w);
    float n2 = w * w;
#pragma unroll
    for (int off = 16; off > 0; off >>= 1) n2 += __shfl_xor(n2, off, 32);
    L = sqrtf(n2);
    v = w / L;
  }

#pragma unroll
  for (int j = 0; j < 32; ++j) ws[lane * 32 + j] = row[j];
  if (lane == 0) ws[1024] = 1.0f / L;
}

// ---------------------------------------------------------------------------
// Kernel 2: per wave, one 32(K) x 16(N) column block.
//   1) AtX tile via WMMA streaming A (L2) and X (HBM, read once).
//   2) 200 FISTA iterations fully in registers; Y re-fed as WMMA B operand
//      via a wave-private LDS layout roundtrip (no barriers needed).
// ---------------------------------------------------------------------------
__global__ __launch_bounds__(256)
void nnls_fista_kernel(const float* __restrict__ X, const float* __restrict__ A,
                       const float* __restrict__ ws, float* __restrict__ out) {
  __shared__ float ybuf[8 * 32 * 16];  // 8 waves x (32 rows x 16 cols)
  const int tid = threadIdx.x;
  const int lane = tid & 31;
  const int wave = tid >> 5;
  const int h = lane >> 4;
  const int cl = lane & 15;
  const int n0 = blockIdx.x * 128 + wave * 16;
  float* yl = &ybuf[wave * 512];

  // ---- AtX = A^T X for this 32x16 tile ----
  v8f xt = {}, xb = {};
  for (int c = 0; c < M_DIM / 4; ++c) {
    const int m0 = 4 * c + 2 * h;
    v2f aT, aB, b;
    aT.x = A[(m0 + 0) * K_DIM + cl];
    aT.y = A[(m0 + 1) * K_DIM + cl];
    aB.x = A[(m0 + 0) * K_DIM + 16 + cl];
    aB.y = A[(m0 + 1) * K_DIM + 16 + cl];
    b.x = X[(m0 + 0) * N_DIM + n0 + cl];
    b.y = X[(m0 + 1) * N_DIM + n0 + cl];
    xt = wmma4(aT, b, xt);
    xb = wmma4(aB, b, xb);
  }

  const float step = ws[1024];
  v8f sxt, sxb;  // step * AtX (constant in the loop)
#pragma unroll
  for (int r = 0; r < 8; ++r) {
    sxt[r] = step * xt[r];
    sxb[r] = step * xb[r];
  }

  // AtA in A-operand layout: element (M=row, K=col) at lane M=cl(+16),
  // chunk c covers cols 4c..4c+3 split across halves/regs.
  v2f a_top[8], a_bot[8];
#pragma unroll
  for (int c = 0; c < 8; ++c) {
    const int k0 = 4 * c + 2 * h;
    a_top[c].x = ws[cl * 32 + k0];
    a_top[c].y = ws[cl * 32 + k0 + 1];
    a_bot[c].x = ws[(16 + cl) * 32 + k0];
    a_bot[c].y = ws[(16 + cl) * 32 + k0 + 1];
  }

  // FISTA state: S, Y in C layout (two 16x16 tiles), Y also in B layout.
  v8f s_t = {}, s_b = {}, y_t = {}, y_b = {};
  float ybl[16];
#pragma unroll
  for (int i = 0; i < 16; ++i) ybl[i] = 0.0f;
  float t = 1.0f;

#pragma unroll 1
  for (int it = 0; it < FISTA_ITERS; ++it) {
    // G = AtA @ Y : 8 chained K=4 WMMAs per output row-tile.
    v8f g_t = {}, g_b = {};
#pragma unroll
    for (int c = 0; c < 8; ++c) {
      v2f b;
      b.x = ybl[2 * c];
      b.y = ybl[2 * c + 1];
      g_t = wmma4(a_top[c], b, g_t);
      g_b = wmma4(a_bot[c], b, g_b);
    }

    const float t_new = 0.5f * (1.0f + sqrtf(1.0f + 4.0f * t * t));
    const float mu = (t - 1.0f) / t_new;
    t = t_new;

#pragma unroll
    for (int r = 0; r < 8; ++r) {
      float snt = fmaxf(fmaf(-step, g_t[r], y_t[r]) + sxt[r], 0.0f);
      y_t[r] = fmaf(mu, snt - s_t[r], snt);
      s_t[r] = snt;
      float snb = fmaxf(fmaf(-step, g_b[r], y_b[r]) + sxb[r], 0.0f);
      y_b[r] = fmaf(mu, snb - s_b[r], snb);
      s_b[r] = snb;
    }

    // C layout -> LDS [row][col] (wave-private; dscnt ordering, no barrier).
#pragma unroll
    for (int r = 0; r < 8; ++r) {
      yl[(r + 8 * h) * 16 + cl]      = y_t[r];
      yl[(16 + r + 8 * h) * 16 + cl] = y_b[r];
    }
    // LDS -> B layout registers for the next iteration.
#pragma unroll
    for (int c = 0; c < 8; ++c) {
      ybl[2 * c]     = yl[(4 * c + 2 * h + 0) * 16 + cl];
      ybl[2 * c + 1] = yl[(4 * c + 2 * h + 1) * 16 + cl];
    }
  }

  // Write S (C layout -> row-major [32][N]).
#pragma unroll
  for (int r = 0; r < 8; ++r) {
    out[(r + 8 * h) * N_DIM + n0 + cl]      = s_t[r];
    out[(16 + r + 8 * h) * N_DIM + n0 + cl] = s_b[r];
  }
}

extern "C" void kernel_launch(void* const* d_in, const int* in_sizes, int n_in,
                              void* d_out, int out_size, void* d_ws,
                              size_t ws_size, hipStream_t stream) {
  (void)in_sizes; (void)n_in; (void)out_size; (void)ws_size;
  const float* X = (const float*)d_in[0];  // [512, 32768]
  const float* A = (const float*)d_in[1];  // [512, 32]
  float* ws = (float*)d_ws;                // [1024] AtA + [1] step
  float* out = (float*)d_out;              // [32, 32768]

  nnls_prep_kernel<<<1, 32, 0, stream>>>(A, ws);
  nnls_fista_kernel<<<N_DIM / 128, 256, 0, stream>>>(X, A, ws, out);
}